// ConvDeepSet_73667279061246
// MI455X (gfx1250) — compile-verified
//
#include <hip/hip_runtime.h>

#define B_     8
#define N_IN   1024
#define N_OUT  4096
#define OC     64
#define BLK    128   // 4 wave32s per block; 256 blocks total for finer WGP spread

typedef float v2f __attribute__((ext_vector_type(2)));
typedef float v8f __attribute__((ext_vector_type(8)));

// out[b,m,o] = sum_c feat[b,m,c]*W[o,c] + bias[o]
//   feat0 = dens = sum_n exp(-0.5*d(n,m)/s0^2)
//   feat1 = (sum_n y[n]*exp(-0.5*d(n,m)/s1^2)) / (dens + 1e-8)
__global__ __launch_bounds__(BLK) void convdeepset_kernel(
    const float* __restrict__ x,      // (B, N_IN, 2)
    const float* __restrict__ y,      // (B, N_IN, 1)
    const float* __restrict__ t,      // (B, N_OUT, 2)
    const float* __restrict__ sigma,  // (2,)
    const float* __restrict__ W,      // (OC, 2)
    const float* __restrict__ bias,   // (OC,)
    float* __restrict__ out)          // (B, N_OUT, OC)
{
    __shared__ float4 sdata[N_IN];    // (x0, x1, y, 0) per n : 16 KB
    __shared__ float  sfeat0[BLK];
    __shared__ float  sfeat1[BLK];

    const int tid = threadIdx.x;
    const int b   = blockIdx.y;
    const int m0  = blockIdx.x * BLK;

    // ---- stage x/y for batch b into LDS (packed for b128 broadcast) ----
    const float* xb = x + (size_t)b * N_IN * 2;
    const float* yb = y + (size_t)b * N_IN;
    for (int n = tid; n < N_IN; n += BLK) {
        float2 xv = ((const float2*)xb)[n];
        float4 s; s.x = xv.x; s.y = xv.y; s.z = yb[n]; s.w = 0.0f;
        sdata[n] = s;
    }

    // exp(-0.5*d/s^2) == exp2(d * coef),  coef = -(0.5/ln2) * exp(-2*sigma)
    const float HALF_OVER_LN2 = 0.7213475204444817f;
    const float LOG2E         = 1.4426950408889634f;
    const float c0 = -HALF_OVER_LN2 * __builtin_amdgcn_exp2f(-2.0f * sigma[0] * LOG2E);
    const float c1 = -HALF_OVER_LN2 * __builtin_amdgcn_exp2f(-2.0f * sigma[1] * LOG2E);

    const int m = m0 + tid;                       // one target point per lane
    const float2 tv = ((const float2*)(t + (size_t)b * N_OUT * 2))[m];

    __syncthreads();

    // ---- phase 1: transcendental-bound reduction over n ----
    float dens = 0.0f, conv = 0.0f;
    #pragma unroll 8
    for (int n = 0; n < N_IN; ++n) {
        float4 s = sdata[n];                      // ds_load_b128, wave broadcast
        float dx = s.x - tv.x;
        float dy = s.y - tv.y;
        float d  = __builtin_fmaf(dy, dy, dx * dx);
        dens += __builtin_amdgcn_exp2f(d * c0);   // v_exp_f32
        conv  = __builtin_fmaf(s.z, __builtin_amdgcn_exp2f(d * c1), conv);
    }

    sfeat0[tid] = dens;
    sfeat1[tid] = conv / (dens + 1e-8f);
    __syncthreads();

    // ---- phase 2: K=2 (padded to 4) channel projection on the matrix pipe ----
    // V_WMMA_F32_16X16X4_F32:  D(16x16) = A(16x4) * B(4x16) + C
    // A layout (32-bit, 16x4): lanes 0-15 -> rows M=0..15, VGPR0=K0, VGPR1=K1;
    //                          lanes 16-31 -> same rows, K2/K3 (zero-padded here).
    // B layout mirrors A with N = lane&15.
    const int lane = tid & 31;
    const int wv   = tid >> 5;          // wave id -> 32 m-rows each
    const int hi   = lane >> 4;         // 0: K0/K1 slots, 1: K2/K3 slots
    const int l15  = lane & 15;

    // Unconditional reads (all lanes use valid addresses), then cndmask the
    // K2/K3 half to zero -- avoids s_and_saveexec divergent-load branches.
    v2f a[2];
    #pragma unroll
    for (int g = 0; g < 2; ++g) {       // two 16-row groups per wave
        const int r = wv * 32 + g * 16 + l15;
        const float f0 = sfeat0[r];
        const float f1 = sfeat1[r];
        a[g].x = hi ? 0.0f : f0;
        a[g].y = hi ? 0.0f : f1;
    }

    #pragma unroll
    for (int ot = 0; ot < 4; ++ot) {    // 4 tiles of 16 output channels
        const int o  = ot * 16 + l15;
        const float w0 = W[o * 2 + 0];  // unconditional, full-EXEC loads
        const float w1 = W[o * 2 + 1];
        const float bv = bias[o];       // C column broadcast = bias
        v2f bm;
        bm.x = hi ? 0.0f : w0;          // B[K=0][N=o]
        bm.y = hi ? 0.0f : w1;          // B[K=1][N=o]

        #pragma unroll
        for (int g = 0; g < 2; ++g) {
            v8f c;
            #pragma unroll
            for (int r = 0; r < 8; ++r) c[r] = bv;

            v8f d = __builtin_amdgcn_wmma_f32_16x16x4_f32(
                false, a[g], false, bm, (short)0, c, false, false);

            // D layout: VGPR r, lanes 0-15 -> M=r, N=lane; lanes 16-31 -> M=8+r.
            const int m_blk = wv * 32 + g * 16 + hi * 8;
            float* op = out + (((size_t)b * N_OUT + (m0 + m_blk)) * OC + o);
            #pragma unroll
            for (int r = 0; r < 8; ++r)
                op[(size_t)r * OC] = d[r];
        }
    }
}

extern "C" void kernel_launch(void* const* d_in, const int* in_sizes, int n_in,
                              void* d_out, int out_size, void* d_ws, size_t ws_size,
                              hipStream_t stream) {
    const float* x     = (const float*)d_in[0];
    const float* y     = (const float*)d_in[1];
    const float* t     = (const float*)d_in[2];
    const float* sigma = (const float*)d_in[3];
    const float* W     = (const float*)d_in[4];
    const float* bias  = (const float*)d_in[5];
    float* out = (float*)d_out;

    dim3 grid(N_OUT / BLK, B_);
    convdeepset_kernel<<<grid, dim3(BLK), 0, stream>>>(x, y, t, sigma, W, bias, out);
}